// SoftL1ACELoss_47777216200804
// MI455X (gfx1250) — compile-verified
//
#include <hip/hip_runtime.h>
#include <stdint.h>

// SoftL1ACELoss for MI455X (gfx1250).
// Strategy: exp factorization (2 transcendentals/elem instead of 20),
// per-lane float2 accumulators for 20 bins x {w, w*p, w*t},
// CDNA5 async global->LDS double-buffered streaming (ASYNCcnt),
// shuffle + LDS + atomic reduction, tiny finalize kernel.

#define NB 20
#define WAVES_PER_BLOCK 8
#define BLOCK_THREADS 256
#define GRID_BLOCKS 1024
#define CHUNK_ELEMS 128            // 32 lanes * 4 floats per chunk per array

// exp(-(p-c)^2/T) = 2^(KA*p^2) * (2^(KB*p))^k * const_k  (const_k cancels in e,o)
#define KA (-14.426950408889634f)  // -log2(e)/T, T=0.1
#define KB (1.5186263588304061f)   // 2*log2(e)/(19*T)

static __device__ __forceinline__ void async_b128(uint32_t lds, uint32_t voff,
                                                  const void* base) {
  // GVS mode: copies 16 bytes per lane from the 64-bit SGPR base plus the 32-bit VGPR offset to the LDS address in vdst; increments ASYNCcnt
  asm volatile("global_load_async_to_lds_b128 %0, %1, %2"
               :: "v"(lds), "v"(voff), "s"(base) : "memory");
}
static __device__ __forceinline__ void wait_async_le2() {
  asm volatile("s_wait_asynccnt 0x2" ::: "memory");
}
static __device__ __forceinline__ void wait_async_le0() {
  asm volatile("s_wait_asynccnt 0x0" ::: "memory");
}

__global__ void __launch_bounds__(BLOCK_THREADS)
ace_partial_kernel(const float* __restrict__ preds,
                   const float* __restrict__ targets,
                   float* __restrict__ ws, int nChunks, int n) {
  __shared__ __align__(16) float sP[WAVES_PER_BLOCK][2][CHUNK_ELEMS];
  __shared__ __align__(16) float sT[WAVES_PER_BLOCK][2][CHUNK_ELEMS];
  __shared__ float blockRed[WAVES_PER_BLOCK][3 * NB];

  const int tid        = blockIdx.x * blockDim.x + threadIdx.x;
  const int lane       = threadIdx.x & 31;
  const int wv         = threadIdx.x >> 5;
  const int gwave      = tid >> 5;
  const int totalWaves = (gridDim.x * blockDim.x) >> 5;

  // Wave-relative LDS byte addresses for this lane's 16B slot in each buffer.
  const uint32_t ldsP[2] = { (uint32_t)(uintptr_t)&sP[wv][0][lane * 4],
                             (uint32_t)(uintptr_t)&sP[wv][1][lane * 4] };
  const uint32_t ldsT[2] = { (uint32_t)(uintptr_t)&sT[wv][0][lane * 4],
                             (uint32_t)(uintptr_t)&sT[wv][1][lane * 4] };

  float2 aw[NB], ap[NB], at[NB];
#pragma unroll
  for (int k = 0; k < NB; ++k) {
    aw[k] = make_float2(0.f, 0.f);
    ap[k] = make_float2(0.f, 0.f);
    at[k] = make_float2(0.f, 0.f);
  }

  // Two elements per call, laid out as float2 lanes to invite v_pk_* packing.
  auto acc2 = [&](float px, float py, float tx, float ty) {
    float ux = __builtin_amdgcn_exp2f(KA * px * px);
    float uy = __builtin_amdgcn_exp2f(KA * py * py);
    float rx = __builtin_amdgcn_exp2f(KB * px);
    float ry = __builtin_amdgcn_exp2f(KB * py);
#pragma unroll
    for (int k = 0; k < NB; ++k) {
      aw[k].x += ux;                       aw[k].y += uy;
      ap[k].x = fmaf(ux, px, ap[k].x);     ap[k].y = fmaf(uy, py, ap[k].y);
      at[k].x = fmaf(ux, tx, at[k].x);     at[k].y = fmaf(uy, ty, at[k].y);
      ux *= rx;                            uy *= ry;
    }
  };

  // ---- async double-buffered streaming over full chunks ----
  int chunk = gwave;
  int buf = 0;
  if (chunk < nChunks) {
    uint32_t off = (uint32_t)chunk * (CHUNK_ELEMS * 4u) + (uint32_t)lane * 16u;
    async_b128(ldsP[0], off, preds);
    async_b128(ldsT[0], off, targets);
  }
  while (chunk < nChunks) {
    const int next = chunk + totalWaves;
    if (next < nChunks) {
      uint32_t off = (uint32_t)next * (CHUNK_ELEMS * 4u) + (uint32_t)lane * 16u;
      async_b128(ldsP[buf ^ 1], off, preds);
      async_b128(ldsT[buf ^ 1], off, targets);
      wait_async_le2();   // current chunk's 2 loads retired (in-order completion)
    } else {
      wait_async_le0();
    }
    const float4 p4 = *reinterpret_cast<const float4*>(&sP[wv][buf][lane * 4]);
    const float4 t4 = *reinterpret_cast<const float4*>(&sT[wv][buf][lane * 4]);
    acc2(p4.x, p4.y, t4.x, t4.y);
    acc2(p4.z, p4.w, t4.z, t4.w);
    buf ^= 1;
    chunk = next;
  }

  // ---- tail (n % CHUNK_ELEMS elements; zero for the harness shape) ----
  const int totalThreads = gridDim.x * blockDim.x;
  for (int i = nChunks * CHUNK_ELEMS + tid; i < n; i += totalThreads) {
    const float pv = preds[i], tv = targets[i];
    float u = __builtin_amdgcn_exp2f(KA * pv * pv);
    const float r = __builtin_amdgcn_exp2f(KB * pv);
#pragma unroll
    for (int k = 0; k < NB; ++k) {
      aw[k].x += u;
      ap[k].x = fmaf(u, pv, ap[k].x);
      at[k].x = fmaf(u, tv, at[k].x);
      u *= r;
    }
  }

  // ---- reduce: pair-combine, wave shuffle, block LDS, global atomics ----
  float red[3 * NB];
#pragma unroll
  for (int k = 0; k < NB; ++k) {
    red[k]          = aw[k].x + aw[k].y;
    red[NB + k]     = ap[k].x + ap[k].y;
    red[2 * NB + k] = at[k].x + at[k].y;
  }
#pragma unroll
  for (int off = 16; off > 0; off >>= 1) {
#pragma unroll
    for (int k = 0; k < 3 * NB; ++k) red[k] += __shfl_down(red[k], off, 32);
  }
  if (lane == 0) {
#pragma unroll
    for (int k = 0; k < 3 * NB; ++k) blockRed[wv][k] = red[k];
  }
  __syncthreads();
  if (threadIdx.x < 3 * NB) {
    float s = 0.f;
#pragma unroll
    for (int w = 0; w < WAVES_PER_BLOCK; ++w) s += blockRed[w][threadIdx.x];
    atomicAdd(&ws[threadIdx.x], s);
  }
}

__global__ void ace_zero_ws(float* __restrict__ ws) {
  if (threadIdx.x < 64) ws[threadIdx.x] = 0.f;
}

__global__ void ace_finalize(const float* __restrict__ ws,
                             float* __restrict__ out) {
  const int k = threadIdx.x;
  float contrib = 0.f;
  if (k < NB) {
    const float w  = ws[k];
    const float wp = ws[NB + k];
    const float wt = ws[2 * NB + k];
    // |e - o| = |Σup - Σut| / (Σu + eps); per-bin constant cancels exactly.
    contrib = (w == 0.f) ? 0.f : fabsf(wp - wt) / (w + 1e-8f);
  }
#pragma unroll
  for (int off = 16; off > 0; off >>= 1) contrib += __shfl_down(contrib, off, 32);
  if (k == 0) out[0] = contrib / (float)NB;
}

extern "C" void kernel_launch(void* const* d_in, const int* in_sizes, int n_in,
                              void* d_out, int out_size, void* d_ws, size_t ws_size,
                              hipStream_t stream) {
  (void)n_in; (void)out_size; (void)ws_size;
  const float* preds   = (const float*)d_in[0];
  const float* targets = (const float*)d_in[1];
  float* ws  = (float*)d_ws;
  float* out = (float*)d_out;
  const int n = in_sizes[0];
  const int nChunks = n / CHUNK_ELEMS;

  ace_zero_ws<<<1, 64, 0, stream>>>(ws);
  ace_partial_kernel<<<GRID_BLOCKS, BLOCK_THREADS, 0, stream>>>(
      preds, targets, ws, nChunks, n);
  ace_finalize<<<1, 32, 0, stream>>>(ws, out);
}